// SimpleAmbienceExtractor_24910810316840
// MI455X (gfx1250) — compile-verified
//
#include <hip/hip_runtime.h>
#include <cstdint>

#define TLEN  262144
#define NFFT  4096
#define HOPS  1024
#define NFRM  257
#define NBINS 2049
#define PADH  2048
#define OLEN  266240   /* NFFT + (NFRM-1)*HOPS */
#define FVEC  8196     /* NBINS*4 */
#define NB    8
#define GN    80       /* padded N for layer-1 GEMM: 5 ctx * 16 */
#define MT    17       /* ceil(257/16) */

typedef __attribute__((ext_vector_type(2))) float v2f;
typedef __attribute__((ext_vector_type(8))) float v8f;

static __device__ __forceinline__ v8f wmma_f32_k4(v2f a, v2f b, v8f c) {
  // D(16x16,f32) += A(16x4,f32) * B(4x16,f32)
  return __builtin_amdgcn_wmma_f32_16x16x4_f32(false, a, false, b, (short)0, c, false, false);
}

static __device__ __forceinline__ float hannw(int s) {
  return 0.5f - 0.5f * __cosf(1.5339808e-3f * (float)s); // 2*pi/4096
}

// digit-reverse base-16, 3 digits: value X[k] of the FFT sits at LDS index rev444(k)
static __device__ __forceinline__ int rev444(int k) {
  return ((k & 15) << 8) | (k & 240) | (k >> 8);
}

// One radix-16 stage over inner span SI (256 -> 16 -> 1). Fully inlined so the
// LDS arrays keep addrspace(3): 32-bit ds addressing, no generic-pointer guards.
template <int SI>
static __device__ __forceinline__ void fft_stage(float (&zr)[NFFT], float (&zi)[NFFT],
                                                 const v2f (&wr)[4], const v2f (&wi)[4],
                                                 const v2f (&wn)[4], float dir,
                                                 int wave, int m16, int hi) {
  __syncthreads();
  const float twb = dir * (6.283185307179586f / (16.0f * (float)SI));
#pragma unroll
  for (int tt = 0; tt < 2; ++tt) {
    const int tile = wave * 2 + tt;       // 16 tiles of 16 columns
    const int col  = tile * 16 + m16;
    const int o    = col / SI;            // SI is a power of 4 -> shifts
    const int ii   = col % SI;
    const int base = o * (SI * 16) + ii;  // addr(m,col) = base + m*SI
    v8f cr = {0.f,0.f,0.f,0.f,0.f,0.f,0.f,0.f};
    v8f ci = {0.f,0.f,0.f,0.f,0.f,0.f,0.f,0.f};
#pragma unroll
    for (int q = 0; q < 4; ++q) {
      const int kb = 4 * q + 2 * hi;      // B layout mirrors A: hi-half holds K+2
      v2f br, bi;
      br[0] = zr[base + kb * SI];         br[1] = zr[base + (kb + 1) * SI];
      bi[0] = zi[base + kb * SI];         bi[1] = zi[base + (kb + 1) * SI];
      cr = wmma_f32_k4(wr[q], br, cr);    // Wr*Xr
      cr = wmma_f32_k4(wn[q], bi, cr);    // -Wi*Xi
      ci = wmma_f32_k4(wr[q], bi, ci);    // Wr*Xi
      ci = wmma_f32_k4(wi[q], br, ci);    // Wi*Xr
    }
    // D layout: lane n=m16, VGPR v -> row v + 8*hi.
    // Twiddle W_{16*SI}^{mr*ii}: geometric in mr -> one sincos + incremental
    // complex rotation (full-rate FMA, no trans-pipe / int->float converts).
    if (SI > 1) {
      float rs, rc;
      __sincosf(twb * (float)ii, &rs, &rc);          // rot = e^{i*twb*ii}
      // rot^8 via three squarings (start angle for lanes with hi==1)
      float c2 = rc * rc - rs * rs, s2 = 2.f * rc * rs;
      float c4 = c2 * c2 - s2 * s2, s4 = 2.f * c2 * s2;
      float c8 = c4 * c4 - s4 * s4, s8 = 2.f * c4 * s4;
      float tc = hi ? c8 : 1.f;
      float ts = hi ? s8 : 0.f;
#pragma unroll
      for (int v = 0; v < 8; ++v) {
        const float xr = cr[v], xi = ci[v];
        zr[base + (v + 8 * hi) * SI] = xr * tc - xi * ts;
        zi[base + (v + 8 * hi) * SI] = xr * ts + xi * tc;
        const float ntc = tc * rc - ts * rs;          // t *= rot
        ts = tc * rs + ts * rc;
        tc = ntc;
      }
    } else {
#pragma unroll
      for (int v = 0; v < 8; ++v) {                   // SI==1: identity twiddle
        zr[base + (v + 8 * hi)] = cr[v];
        zi[base + (v + 8 * hi)] = ci[v];
      }
    }
  }
}

// In-LDS 4096-pt complex FFT, radix 16^3, WMMA f32 16x16x4 stages.
// dir = -1 forward, +1 inverse (unscaled). blockDim.x must be 256 (8 waves).
static __device__ __forceinline__ void fft4096_wmma(float (&zr)[NFFT], float (&zi)[NFFT],
                                                    float dir, int tid) {
  const int lane = tid & 31;
  const int wave = tid >> 5;
  const int m16  = lane & 15;
  const int hi   = lane >> 4;

  // A fragments: W16[m][k] = e^{dir*2pi*i*m*k/16}, 16x16 split into 4 K-chunks of 4.
  // A layout: lanes 0-15 -> K base 0/1, lanes 16-31 -> K base 2/3.
  v2f wr[4], wi[4], wn[4];
  const float a16 = dir * 0.39269908169872414f; // dir * 2*pi/16
#pragma unroll
  for (int q = 0; q < 4; ++q) {
    const int k0 = 4 * q + 2 * hi;
    const int p0 = (m16 * k0) & 15;
    const int p1 = (m16 * (k0 + 1)) & 15;
    float s0, c0, s1, c1;
    __sincosf(a16 * (float)p0, &s0, &c0);
    __sincosf(a16 * (float)p1, &s1, &c1);
    wr[q][0] = c0;  wr[q][1] = c1;
    wi[q][0] = s0;  wi[q][1] = s1;
    wn[q][0] = -s0; wn[q][1] = -s1;
  }

  fft_stage<256>(zr, zi, wr, wi, wn, dir, wave, m16, hi);
  fft_stage<16 >(zr, zi, wr, wi, wn, dir, wave, m16, hi);
  fft_stage<1  >(zr, zi, wr, wi, wn, dir, wave, m16, hi);
  __syncthreads();
}

// ---------------- STFT: window + packed complex FFT + Hermitian unpack ------
__global__ void __launch_bounds__(256) k_stft(const float* __restrict__ x,
                                              float* __restrict__ feats) {
  __shared__ float zr[NFFT];
  __shared__ float zi[NFFT];
  const int t = blockIdx.x, b = blockIdx.y;
  const int tid = threadIdx.x;
  const float* xl = x + ((size_t)b * 2 + 0) * TLEN;
  const float* xr = x + ((size_t)b * 2 + 1) * TLEN;
  const int s0 = t * HOPS - PADH;
#pragma unroll
  for (int r = 0; r < 16; ++r) {
    const int s = tid + (r << 8);
    int j = s0 + s;                       // reflect padding
    j = (j < 0) ? -j : j;
    j = (j >= TLEN) ? (2 * TLEN - 2 - j) : j;
    const float w = hannw(s);
    zr[s] = xl[j] * w;                    // z = l + i*r (one FFT, two channels)
    zi[s] = xr[j] * w;
  }
  fft4096_wmma(zr, zi, -1.0f, tid);
  float* fo = feats + ((size_t)b * NFRM + t) * FVEC;
  for (int k = tid; k < NBINS; k += 256) {
    const int km = (NFFT - k) & (NFFT - 1);
    const int pk = rev444(k), pm = rev444(km);
    const float ar = zr[pk], ai = zi[pk];
    const float br = zr[pm], bi = zi[pm];
    fo[k * 4 + 0] = 0.5f * (ar + br);     // L.re
    fo[k * 4 + 1] = 0.5f * (ai + bi);     // R.re
    fo[k * 4 + 2] = 0.5f * (ai - bi);     // L.im
    fo[k * 4 + 3] = 0.5f * (br - ar);     // R.im
  }
}

// ---------------- repack w1 (40980x15) -> G (8196x80) -----------------------
__global__ void k_repack_w1(const float* __restrict__ w1, float* __restrict__ G, int n) {
  const int gid = blockIdx.x * 256 + threadIdx.x;
  if (gid >= n) return;
  const int row  = gid / GN;              // bin*4 + f
  const int colj = gid - row * GN;        // c*16 + j
  const int c = colj >> 4, j = colj & 15;
  const int bin = row >> 2, f = row & 3;
  float v = 0.f;
  if (j < 15) v = w1[(size_t)(bin * 20 + c * 4 + f) * 15 + j];
  G[gid] = v;
}

// ---------------- layer-1 GEMM: P(257x80) = feats(257x8196) @ G(8196x80) ----
__global__ void __launch_bounds__(32) k_gemm(const float* __restrict__ feats,
                                             const float* __restrict__ G,
                                             float* __restrict__ P) {
  const int mt = blockIdx.x, b = blockIdx.y;
  const int lane = threadIdx.x;
  const int m16 = lane & 15, hi = lane >> 4;
  const int tA = mt * 16 + m16;
  const bool tin = tA < NFRM;
  const float* arow = feats + ((size_t)b * NFRM + (tin ? tA : 0)) * FVEC;
  v8f acc[5];
#pragma unroll
  for (int nt = 0; nt < 5; ++nt) {
    v8f z = {0.f,0.f,0.f,0.f,0.f,0.f,0.f,0.f};
    acc[nt] = z;
  }
  for (int ks = 0; ks < FVEC / 4; ++ks) {     // K = 8196, 4 per wmma
    const int ka = ks * 4 + 2 * hi;
    if ((ks & 127) == 0) {
      __builtin_prefetch(arow + ks * 4 + 2048, 0, 1);
      __builtin_prefetch(G + (size_t)(ks * 4 + 2048) * GN + m16, 0, 1);
    }
    v2f a;
    a[0] = tin ? arow[ka] : 0.f;
    a[1] = tin ? arow[ka + 1] : 0.f;
    const float* g0 = G + (size_t)ka * GN;
#pragma unroll
    for (int nt = 0; nt < 5; ++nt) {
      v2f bb;
      bb[0] = g0[nt * 16 + m16];
      bb[1] = g0[GN + nt * 16 + m16];
      acc[nt] = wmma_f32_k4(a, bb, acc[nt]);
    }
  }
#pragma unroll
  for (int nt = 0; nt < 5; ++nt) {
#pragma unroll
    for (int v = 0; v < 8; ++v) {
      const int mr = v + 8 * hi;
      const int tt = mt * 16 + mr;
      if (tt < NFRM)
        P[((size_t)b * NFRM + tt) * GN + nt * 16 + m16] = acc[nt][v];
    }
  }
}

// ---------------- context sum + layers 2/3 + sigmoid -> per-frame mask ------
__global__ void k_mlp_mask(const float* __restrict__ P,
                           const float* __restrict__ b1,
                           const float* __restrict__ w2, const float* __restrict__ b2,
                           const float* __restrict__ w3, const float* __restrict__ b3,
                           float* __restrict__ mask) {
  const int gid = blockIdx.x * 256 + threadIdx.x;
  if (gid >= NB * NFRM) return;
  const int b = gid / NFRM, t = gid - b * NFRM;
  float h1[15];
#pragma unroll
  for (int j = 0; j < 15; ++j) {
    float s = b1[j];
#pragma unroll
    for (int c = 0; c < 5; ++c) {
      const int tc = t + c - 2;
      if (tc >= 0 && tc < NFRM)
        s += P[((size_t)b * NFRM + tc) * GN + c * 16 + j];
    }
    h1[j] = fmaxf(s, 0.f);
  }
  float h2[10];
#pragma unroll
  for (int k = 0; k < 10; ++k) {
    float s = b2[k];
#pragma unroll
    for (int j = 0; j < 15; ++j) s += h1[j] * w2[j * 10 + k];
    h2[k] = fmaxf(s, 0.f);
  }
  float s = b3[0];
#pragma unroll
  for (int k = 0; k < 10; ++k) s += h2[k] * w3[k];
  mask[gid] = 1.0f / (1.0f + __expf(-s));
}

// ---------------- iSTFT: mask, Hermitian rebuild, inverse FFT, OLA ----------
__global__ void __launch_bounds__(256) k_istft(const float* __restrict__ feats,
                                               const float* __restrict__ mask,
                                               float* __restrict__ ola) {
  __shared__ float zr[NFFT];
  __shared__ float zi[NFFT];
  const int t = blockIdx.x, b = blockIdx.y;
  const int tid = threadIdx.x;
  const float m = mask[b * NFRM + t];
  const float* fi = feats + ((size_t)b * NFRM + t) * FVEC;
#pragma unroll
  for (int r = 0; r < 16; ++r) {
    const int k = tid + (r << 8);
    const int km = (k <= 2048) ? k : (NFFT - k);
    const float lre = fi[km * 4 + 0], rre = fi[km * 4 + 1];
    const float lim = fi[km * 4 + 2], rim = fi[km * 4 + 3];
    float zre, zim;
    if (k <= 2048) { zre = m * (lre - rim); zim = m * (lim + rre); }         // Z = Sl + i*Sr
    else           { zre = m * (lre + rim); zim = m * (rre - lim); }         // conj extension
    zr[k] = zre; zi[k] = zim;
  }
  fft4096_wmma(zr, zi, 1.0f, tid);
  float* oL = ola + ((size_t)(b * 2 + 0)) * OLEN + t * HOPS;
  float* oR = ola + ((size_t)(b * 2 + 1)) * OLEN + t * HOPS;
  const float sc = 1.0f / (float)NFFT;   // irfft 1/N
#pragma unroll
  for (int r = 0; r < 16; ++r) {
    const int s = tid + (r << 8);
    const int p = rev444(s);
    const float w = hannw(s) * sc;
    atomicAdd(&oL[s], zr[p] * w);        // l = Re, r = Im
    atomicAdd(&oR[s], zi[p] * w);
  }
}

// ---------------- helpers: zero OLA, final normalize ------------------------
__global__ void k_zero(float* __restrict__ p, int n) {
  const int gid = blockIdx.x * 256 + threadIdx.x;
  if (gid < n) p[gid] = 0.f;
}

__global__ void k_norm(const float* __restrict__ ola, float* __restrict__ out) {
  const int gid = blockIdx.x * 256 + threadIdx.x; // NB*2*TLEN threads exactly
  const int i  = gid & (TLEN - 1);
  const int bc = gid >> 18;                       // TLEN = 2^18
  const int pos = i + PADH;
  int tmax = pos >> 10; if (tmax > NFRM - 1) tmax = NFRM - 1;
  const int tmin = (pos >= NFFT) ? ((pos - (NFFT - 1) + (HOPS - 1)) >> 10) : 0;
  float env = 0.f;
  for (int tt = tmin; tt <= tmax; ++tt) {         // <= 4 overlapping frames
    const float w = hannw(pos - (tt << 10));
    env += w * w;
  }
  const float den = (env > 1e-11f) ? env : 1.0f;
  out[gid] = ola[(size_t)bc * OLEN + pos] / den;
}

extern "C" void kernel_launch(void* const* d_in, const int* in_sizes, int n_in,
                              void* d_out, int out_size, void* d_ws, size_t ws_size,
                              hipStream_t stream) {
  (void)in_sizes; (void)n_in; (void)out_size; (void)ws_size;
  const float* x  = (const float*)d_in[0];
  const float* w1 = (const float*)d_in[1];
  const float* b1 = (const float*)d_in[2];
  const float* w2 = (const float*)d_in[3];
  const float* b2 = (const float*)d_in[4];
  const float* w3 = (const float*)d_in[5];
  const float* b3 = (const float*)d_in[6];
  float* out = (float*)d_out;
  float* ws  = (float*)d_ws;

  const size_t nFeats = (size_t)NB * NFRM * FVEC;   // 16,850,976 f32 (67 MB)
  const size_t nG     = (size_t)FVEC * GN;          //    655,680
  const size_t nP     = (size_t)NB * NFRM * GN;     //    164,480
  const size_t nMask  = (size_t)NB * NFRM;          //      2,056
  const size_t nOla   = (size_t)NB * 2 * OLEN;      //  4,259,840 (17 MB)

  float* feats = ws;
  float* G     = feats + ((nFeats + 63) & ~(size_t)63);
  float* P     = G     + ((nG     + 63) & ~(size_t)63);
  float* mask  = P     + ((nP     + 63) & ~(size_t)63);
  float* ola   = mask  + ((nMask  + 63) & ~(size_t)63);

  k_zero<<<(int)((nOla + 255) / 256), 256, 0, stream>>>(ola, (int)nOla);
  k_repack_w1<<<(int)((nG + 255) / 256), 256, 0, stream>>>(w1, G, (int)nG);
  k_stft<<<dim3(NFRM, NB), 256, 0, stream>>>(x, feats);
  k_gemm<<<dim3(MT, NB), 32, 0, stream>>>(feats, G, P);
  k_mlp_mask<<<(int)((nMask + 255) / 256), 256, 0, stream>>>(P, b1, w2, b2, w3, b3, mask);
  k_istft<<<dim3(NFRM, NB), 256, 0, stream>>>(feats, mask, ola);
  k_norm<<<16384, 256, 0, stream>>>(ola, out);
}